// WHVILinear_38517266710582
// MI455X (gfx1250) — compile-verified
//
#include <hip/hip_runtime.h>

// WHVI: y = s1 * FWHT( u * FWHT( s2 * x ) ),  D = 4096 = 16^3, B = 16384.
// FWHT-4096 == (H16 (x) H16 (x) H16): three rounds of 16x16 constant +-1
// matrix products -> V_WMMA_F32_16X16X4_F32 (4 chained K=4 WMMAs per tile).
// One wave owns one sample row staged in a private 16KB LDS slab.

typedef float v2f __attribute__((ext_vector_type(2)));
typedef float v8f __attribute__((ext_vector_type(8)));

#ifndef WHVI_DIM
#define WHVI_DIM 4096
#endif
#define WAVES_PER_BLOCK 4
#define THREADS_PER_BLOCK (WAVES_PER_BLOCK * 32)

__device__ __forceinline__ float hsign(int r, int c) {
    // Sylvester Hadamard entry: (-1)^popcount(r & c)
    return (__popc(r & c) & 1) ? -1.0f : 1.0f;
}

// One full 16x16x16-tensor Hadamard transform (= FWHT-4096) in LDS.
// lds: 4096 floats, index d = i*256 + j*16 + k.
// hf[c]: per-lane H16 fragment for K-chunk c (valid as both A and B operand
// of V_WMMA_F32_16X16X4_F32 since H16 is symmetric).
__device__ __forceinline__ void fwht4096_lds(float* lds, const v2f hf[4],
                                             int half, int m) {
    // ---- Stage A: contract k.  Row-tiles T_t (rows 16t..16t+15, cols k).
    // D = T_t * H   (A = data, B = H)
    for (int t = 0; t < 16; ++t) {
        v8f acc = {};
#pragma unroll
        for (int c = 0; c < 4; ++c) {
            const int k0 = 4 * c + 2 * half;
            v2f a;
            a.x = lds[(16 * t + m) * 16 + k0];
            a.y = lds[(16 * t + m) * 16 + k0 + 1];
            acc = __builtin_amdgcn_wmma_f32_16x16x4_f32(
                false, a, false, hf[c], (short)0, acc, false, false);
        }
#pragma unroll
        for (int g = 0; g < 8; ++g)
            lds[(16 * t + g + 8 * half) * 16 + m] = acc[g];
    }
    // ---- Stage B: contract j.  For each slice i: D = H * S_i.
    for (int i = 0; i < 16; ++i) {
        v8f acc = {};
#pragma unroll
        for (int c = 0; c < 4; ++c) {
            const int j0 = 4 * c + 2 * half;
            v2f b;
            b.x = lds[i * 256 + j0 * 16 + m];
            b.y = lds[i * 256 + (j0 + 1) * 16 + m];
            acc = __builtin_amdgcn_wmma_f32_16x16x4_f32(
                false, hf[c], false, b, (short)0, acc, false, false);
        }
#pragma unroll
        for (int g = 0; g < 8; ++g)
            lds[i * 256 + (g + 8 * half) * 16 + m] = acc[g];
    }
    // ---- Stage C: contract i.  Column-tiles over (j,k): D = H * W_t.
    for (int t = 0; t < 16; ++t) {
        v8f acc = {};
#pragma unroll
        for (int c = 0; c < 4; ++c) {
            const int i0 = 4 * c + 2 * half;
            v2f b;
            b.x = lds[i0 * 256 + t * 16 + m];
            b.y = lds[(i0 + 1) * 256 + t * 16 + m];
            acc = __builtin_amdgcn_wmma_f32_16x16x4_f32(
                false, hf[c], false, b, (short)0, acc, false, false);
        }
#pragma unroll
        for (int g = 0; g < 8; ++g)
            lds[(g + 8 * half) * 256 + t * 16 + m] = acc[g];
    }
}

__global__ __launch_bounds__(THREADS_PER_BLOCK)
void whvi_kernel(const float* __restrict__ x,
                 const float* __restrict__ epsilon,
                 const float* __restrict__ s1,
                 const float* __restrict__ s2,
                 const float* __restrict__ g_mu,
                 const float* __restrict__ g_rho,
                 float* __restrict__ out) {
    __shared__ float smem[WAVES_PER_BLOCK * WHVI_DIM];  // 64 KB

    const int lane = threadIdx.x & 31;
    const int wave = threadIdx.x >> 5;
    const int half = lane >> 4;   // 0: lanes 0-15, 1: lanes 16-31
    const int m    = lane & 15;

    float* lds = smem + wave * WHVI_DIM;
    const long sample = (long)blockIdx.x * WAVES_PER_BLOCK + wave;

    const float4* __restrict__ xrow = (const float4*)(x + sample * WHVI_DIM);
    float4* __restrict__       orow = (float4*)(out + sample * WHVI_DIM);
    const float4* __restrict__ s1v  = (const float4*)s1;
    const float4* __restrict__ s2v  = (const float4*)s2;
    const float4* __restrict__ muv  = (const float4*)g_mu;
    const float4* __restrict__ rhov = (const float4*)g_rho;
    const float4* __restrict__ epsv = (const float4*)epsilon;
    float4* __restrict__ lds4 = (float4*)lds;

    // Per-lane H16 fragments (A- and B-operand compatible; H16 symmetric).
    v2f hf[4];
#pragma unroll
    for (int c = 0; c < 4; ++c) {
        const int k0 = 4 * c + 2 * half;
        hf[c].x = hsign(k0, m);
        hf[c].y = hsign(k0 + 1, m);
    }

    constexpr int VEC_ITERS = WHVI_DIM / 4 / 32;  // 32 float4 per lane

    // Pass 1: LDS <- x * s2   (s2/u/s1 vectors are 16KB, L2-resident)
#pragma unroll 4
    for (int it = 0; it < VEC_ITERS; ++it) {
        const int idx = it * 32 + lane;
        const float4 xv = xrow[idx];
        const float4 sv = s2v[idx];
        float4 r;
        r.x = xv.x * sv.x; r.y = xv.y * sv.y;
        r.z = xv.z * sv.z; r.w = xv.w * sv.w;
        lds4[idx] = r;
    }

    fwht4096_lds(lds, hf, half, m);

    // Pass 2: LDS *= u,  u = g_mu + softplus(g_rho) * eps
#pragma unroll 4
    for (int it = 0; it < VEC_ITERS; ++it) {
        const int idx = it * 32 + lane;
        const float4 mu = muv[idx];
        const float4 rh = rhov[idx];
        const float4 ep = epsv[idx];
        float4 v = lds4[idx];
        v.x *= mu.x + log1pf(__expf(rh.x)) * ep.x;
        v.y *= mu.y + log1pf(__expf(rh.y)) * ep.y;
        v.z *= mu.z + log1pf(__expf(rh.z)) * ep.z;
        v.w *= mu.w + log1pf(__expf(rh.w)) * ep.w;
        lds4[idx] = v;
    }

    fwht4096_lds(lds, hf, half, m);

    // Pass 3: out <- LDS * s1
#pragma unroll 4
    for (int it = 0; it < VEC_ITERS; ++it) {
        const int idx = it * 32 + lane;
        const float4 sv = s1v[idx];
        float4 v = lds4[idx];
        v.x *= sv.x; v.y *= sv.y; v.z *= sv.z; v.w *= sv.w;
        orow[idx] = v;
    }
}

extern "C" void kernel_launch(void* const* d_in, const int* in_sizes, int n_in,
                              void* d_out, int out_size, void* d_ws, size_t ws_size,
                              hipStream_t stream) {
    const float* x    = (const float*)d_in[0];
    const float* eps  = (const float*)d_in[1];
    const float* s1   = (const float*)d_in[2];
    const float* s2   = (const float*)d_in[3];
    const float* g_mu = (const float*)d_in[4];
    const float* g_rho= (const float*)d_in[5];
    float* out = (float*)d_out;

    const int rows = in_sizes[0] / WHVI_DIM;          // 16384
    const int blocks = rows / WAVES_PER_BLOCK;        // 4096
    whvi_kernel<<<blocks, THREADS_PER_BLOCK, 0, stream>>>(
        x, eps, s1, s2, g_mu, g_rho, out);
}